// InnerProduct_65429531787441
// MI455X (gfx1250) — compile-verified
//
#include <hip/hip_runtime.h>

#define FDIM 50
#define EDIM 64
#define DDIM 128
#define KP   52      // F padded to multiple of 4
#define NKK  13      // KP / 4 k-steps per GEMM
#define BLOCK 256

typedef float v2f __attribute__((ext_vector_type(2)));
typedef float v8f __attribute__((ext_vector_type(8)));
typedef int   v4i __attribute__((ext_vector_type(4)));

// Pointee type must be int4; AS1 = global ("__device__"), AS3 = LDS.
typedef __attribute__((address_space(1))) v4i gv4i;
typedef __attribute__((address_space(3))) v4i lv4i;

// ---- async global -> LDS copy of 16 bytes (per-lane), ASYNCcnt-tracked ----
__device__ __forceinline__ void async_copy16(const float* g, float* l) {
#if __has_builtin(__builtin_amdgcn_global_load_async_to_lds_b128)
  __builtin_amdgcn_global_load_async_to_lds_b128(
      (gv4i*)(unsigned long long)(const void*)g,
      (lv4i*)(unsigned int)(unsigned long long)(const void*)l,
      0, 0);
#else
  float4 t = *(const float4*)g;
  *(float4*)l = t;
#endif
}

__device__ __forceinline__ void wait_async0() {
#if __has_builtin(__builtin_amdgcn_s_wait_asynccnt)
  __builtin_amdgcn_s_wait_asynccnt(0);
#else
  asm volatile("s_wait_asynccnt 0" ::: "memory");
#endif
}

__global__ __launch_bounds__(BLOCK)
void InnerProduct_65429531787441_kernel(const float* __restrict__ X,
                                        const float* __restrict__ Theta,
                                        float* __restrict__ out,
                                        int Btot) {
  __shared__ float sTh[DDIM * KP];          // 128 x 52 f32, zero-padded K
  __shared__ float sX[2][KP * EDIM];        // double-buffered 52 x 64 f32

  const int tid    = threadIdx.x;
  const int wave   = tid >> 5;              // 0..7  -> M-tile of D
  const int lane   = tid & 31;
  const int laneLo = lane & 15;
  const int hi     = lane >> 4;             // 0: K 0/1, rows M=v ; 1: K 2/3, rows M=v+8

  // ---- stage Theta into LDS (zero-pad K from 50 to 52) ----
  for (int idx = tid; idx < DDIM * KP; idx += BLOCK) {
    int d = idx / KP;
    int f = idx - d * KP;
    sTh[idx] = (f < FDIM) ? Theta[d * FDIM + f] : 0.0f;
  }
  // ---- zero the pad rows (50,51) of both X buffers once ----
  for (int idx = tid; idx < 2 * 2 * EDIM; idx += BLOCK) {
    int bufi = idx >> 7;
    sX[bufi][FDIM * EDIM + (idx & 127)] = 0.0f;
  }

  const int gstride = gridDim.x;
  const int b0 = blockIdx.x;

  // prologue: async-load first batch into buffer 0
  if (b0 < Btot) {
    const float* src = X + (size_t)b0 * (FDIM * EDIM);
    for (int q = tid; q < (FDIM * EDIM) / 4; q += BLOCK)
      async_copy16(src + 4 * q, &sX[0][4 * q]);
  }

  int buf = 0;
  for (int b = b0; b < Btot; b += gstride) {
    wait_async0();        // my async copies into sX[buf] done
    __syncthreads();      // everyone's copies visible; prev compute finished

    const int nb = b + gstride;
    if (nb < Btot) {      // overlap next batch load with compute
      const float* src = X + (size_t)nb * (FDIM * EDIM);
      float* dst = &sX[buf ^ 1][0];
      for (int q = tid; q < (FDIM * EDIM) / 4; q += BLOCK)
        async_copy16(src + 4 * q, dst + 4 * q);
    }

    // ---- GEMM tile: D-rows [wave*16, wave*16+16) x full E=64 ----
    const float* xb = &sX[buf][0];
    const float* aRowPtr = &sTh[(wave * 16 + laneLo) * KP + 2 * hi];

    v8f acc[4] = {};
#pragma unroll
    for (int kk = 0; kk < NKK; ++kk) {
      const int kb = 4 * kk + 2 * hi;
      // A fragment: K = kb, kb+1 contiguous in LDS -> one b64 load
      v2f a = *(const v2f*)(aRowPtr + 4 * kk);
#pragma unroll
      for (int n = 0; n < 4; ++n) {
        v2f bf;
        bf.x = xb[kb * EDIM + n * 16 + laneLo];
        bf.y = xb[(kb + 1) * EDIM + n * 16 + laneLo];
        acc[n] = __builtin_amdgcn_wmma_f32_16x16x4_f32(
            /*neg_a=*/false, a, /*neg_b=*/false, bf,
            /*c_mod=*/(short)0, acc[n],
            /*reuse_a=*/false, /*reuse_b=*/false);
      }
    }

    // ---- sum of squares over E: per-lane partial, then 16-lane butterfly ----
    float s[8];
#pragma unroll
    for (int v = 0; v < 8; ++v) {
      float t = 0.0f;
#pragma unroll
      for (int n = 0; n < 4; ++n) {
        float c = acc[n][v];
        t += c * c;
      }
      s[v] = t;
    }
#pragma unroll
    for (int v = 0; v < 8; ++v) {
      s[v] += __shfl_xor(s[v], 1, 32);
      s[v] += __shfl_xor(s[v], 2, 32);
      s[v] += __shfl_xor(s[v], 4, 32);
      s[v] += __shfl_xor(s[v], 8, 32);
    }
    // C layout: VGPR v, lane<16 -> M=v, lane>=16 -> M=v+8
    if (laneLo == 0) {
      float* o = out + (size_t)b * DDIM + wave * 16 + 8 * hi;
#pragma unroll
      for (int v = 0; v < 8; ++v) o[v] = s[v];
    }

    buf ^= 1;
  }
}

extern "C" void kernel_launch(void* const* d_in, const int* in_sizes, int n_in,
                              void* d_out, int out_size, void* d_ws, size_t ws_size,
                              hipStream_t stream) {
  const float* X     = (const float*)d_in[0];   // [B, 50, 64] f32
  const float* Theta = (const float*)d_in[1];   // [128, 50]   f32
  float* out = (float*)d_out;                   // [B, 128]    f32

  int Btot = in_sizes[0] / (FDIM * EDIM);
  int blocks = 2048;
  if (blocks > Btot) blocks = Btot;
  if (blocks < 1) blocks = 1;

  InnerProduct_65429531787441_kernel<<<blocks, BLOCK, 0, stream>>>(X, Theta, out, Btot);
}